// MPMModelLearnedPhi_70944269796238
// MI455X (gfx1250) — compile-verified
//
#include <hip/hip_runtime.h>
#include <hip/hip_bf16.h>

#define NPM_THREADS 256

namespace {
constexpr int   kGrid    = 256;
constexpr int   kCells   = kGrid * kGrid;
constexpr float kDX      = (float)(1.0 / 256.0);
constexpr float kInvDX   = 256.0f;
constexpr float kDT      = (float)1.0e-4;
constexpr float kPVol    = (float)((0.5 / 256.0) * (0.5 / 256.0));
constexpr float kPMass   = kPVol;                                   // rho = 1
constexpr float kMu      = (float)(1000.0 / (2.0 * (1.0 + 0.2)));   // 416.666...
constexpr float kLa      = (float)(1000.0 * 0.2 / ((1.0 + 0.2) * (1.0 - 0.4)));
constexpr float kEps     = 1.0e-8f;
// -DT * P_VOL * 4 * INV_DX^2 == -1e-4 exactly (powers of two cancel)
constexpr float kStressC = (float)(-1.0e-4 * ((0.5/256.0)*(0.5/256.0)) * 4.0 * 256.0 * 256.0);
constexpr float kScaleC  = (float)(4.0 * 256.0 * 256.0);            // 262144
constexpr float kDTG     = (float)(1.0e-4 * 9.8);
constexpr float kXlo     = (float)(0.51 / 256.0);
constexpr float kXhi     = (float)((256.0 - 1.51) / 256.0);
}

typedef float v2f __attribute__((ext_vector_type(2)));
typedef float v4f __attribute__((ext_vector_type(4)));

// Hardware f32 atomic add (global_atomic_add_f32, no-return -> STOREcnt path)
__device__ __forceinline__ void fatomic_add(float* p, float v) {
  unsafeAtomicAdd(p, v);
}

// CDNA5 async global->LDS staging (ASYNCcnt path), guarded so compile never breaks.
#if __has_builtin(__builtin_amdgcn_global_load_async_to_lds_b64) && \
    __has_builtin(__builtin_amdgcn_global_load_async_to_lds_b128)
#define MPM_ASYNC_LDS 1
typedef int v2i_t __attribute__((vector_size(8)));
typedef int v4i_t __attribute__((vector_size(16)));
typedef __attribute__((address_space(1))) v2i_t GV2;   // global v2i
typedef __attribute__((address_space(3))) v2i_t LV2;   // LDS    v2i
typedef __attribute__((address_space(1))) v4i_t GV4;   // global v4i
typedef __attribute__((address_space(3))) v4i_t LV4;   // LDS    v4i

__device__ __forceinline__ void wait_async0() {
#if __has_builtin(__builtin_amdgcn_s_wait_asynccnt)
  __builtin_amdgcn_s_wait_asynccnt(0);
#else
  asm volatile("s_wait_asynccnt 0x0" ::: "memory");
#endif
}
#else
#define MPM_ASYNC_LDS 0
#endif

__global__ __launch_bounds__(NPM_THREADS) void k_zero4(v4f* __restrict__ g, int count4) {
  int i = blockIdx.x * NPM_THREADS + threadIdx.x;
  if (i < count4) g[i] = (v4f){0.0f, 0.0f, 0.0f, 0.0f};
}

__global__ __launch_bounds__(NPM_THREADS) void k_p2g(
    const float* __restrict__ xg, const float* __restrict__ vg,
    const float* __restrict__ Cg, const float* __restrict__ Fg,
    const float* __restrict__ Wg,
    const int*   __restrict__ matg, const float* __restrict__ Jpg,
    float* __restrict__ grid_v, float* __restrict__ grid_m,
    float* __restrict__ out_F, int* __restrict__ out_mat, float* __restrict__ out_Jp,
    int n) {
  const int t = threadIdx.x;
  const int p = blockIdx.x * NPM_THREADS + t;

#if MPM_ASYNC_LDS
  __shared__ float sX[2 * NPM_THREADS];
  __shared__ float sV[2 * NPM_THREADS];
  __shared__ float sC[4 * NPM_THREADS];
  __shared__ float sF[4 * NPM_THREADS];
  if (p < n) {
    __builtin_amdgcn_global_load_async_to_lds_b64 ((GV2*)(xg + 2 * (size_t)p), (LV2*)(sX + 2 * t), 0, 0);
    __builtin_amdgcn_global_load_async_to_lds_b64 ((GV2*)(vg + 2 * (size_t)p), (LV2*)(sV + 2 * t), 0, 0);
    __builtin_amdgcn_global_load_async_to_lds_b128((GV4*)(Cg + 4 * (size_t)p), (LV4*)(sC + 4 * t), 0, 0);
    __builtin_amdgcn_global_load_async_to_lds_b128((GV4*)(Fg + 4 * (size_t)p), (LV4*)(sF + 4 * t), 0, 0);
  }
  wait_async0();
  if (p >= n) return;
  const float x0 = sX[2*t+0], x1 = sX[2*t+1];
  const float v0 = sV[2*t+0], v1 = sV[2*t+1];
  const float C00 = sC[4*t+0], C01 = sC[4*t+1], C10 = sC[4*t+2], C11 = sC[4*t+3];
  const float F00 = sF[4*t+0], F01 = sF[4*t+1], F10 = sF[4*t+2], F11 = sF[4*t+3];
#else
  if (p >= n) return;
  const float x0 = xg[2*p+0], x1 = xg[2*p+1];
  const float v0 = vg[2*p+0], v1 = vg[2*p+1];
  const float C00 = Cg[4*p+0], C01 = Cg[4*p+1], C10 = Cg[4*p+2], C11 = Cg[4*p+3];
  const float F00 = Fg[4*p+0], F01 = Fg[4*p+1], F10 = Fg[4*p+2], F11 = Fg[4*p+3];
#endif

  const float W0 = Wg[0], W1 = Wg[1];

  const int   b0  = (int)(x0 * kInvDX - 0.5f);
  const int   b1  = (int)(x1 * kInvDX - 0.5f);
  const float fx0 = x0 * kInvDX - (float)b0;
  const float fx1 = x1 * kInvDX - (float)b1;

  float wa[3], wb[3];
  wa[0] = 0.5f * (1.5f - fx0) * (1.5f - fx0);
  wa[1] = 0.75f - (fx0 - 1.0f) * (fx0 - 1.0f);
  wa[2] = 0.5f * (fx0 - 0.5f) * (fx0 - 0.5f);
  wb[0] = 0.5f * (1.5f - fx1) * (1.5f - fx1);
  wb[1] = 0.75f - (fx1 - 1.0f) * (fx1 - 1.0f);
  wb[2] = 0.5f * (fx1 - 0.5f) * (fx1 - 0.5f);

  // F_new = F + DT * (C @ F)
  const float Fn00 = F00 + kDT * (C00 * F00 + C01 * F10);
  const float Fn01 = F01 + kDT * (C00 * F01 + C01 * F11);
  const float Fn10 = F10 + kDT * (C10 * F00 + C11 * F10);
  const float Fn11 = F11 + kDT * (C10 * F01 + C11 * F11);

  // ---- analytic grad of psi_sum w.r.t. F_new (replicates JAX autodiff chain) ----
  const float c00 = Fn00 * Fn00 + Fn10 * Fn10;   // C = F^T F
  const float c11 = Fn01 * Fn01 + Fn11 * Fn11;
  const float c01 = Fn00 * Fn01 + Fn10 * Fn11;
  const float tr  = c00 + c11;
  const float det = c00 * c11 - c01 * c01;
  const float d2  = tr * tr - 4.0f * det;
  const float delta = sqrtf(fmaxf(d2, kEps));
  const float a1 = 0.5f * (tr + delta);
  const float a2 = 0.5f * (tr - delta);
  const float s1 = sqrtf(fmaxf(a1, kEps));
  const float s2 = sqrtf(fmaxf(a2, kEps));

  const float prod = s1 * s2 - 1.0f;
  const float g1 = 2.0f * kMu * (s1 - 1.0f) + kLa * prod * s2 + W0;  // dpsi/ds1
  const float g2 = 2.0f * kMu * (s2 - 1.0f) + kLa * prod * s1 + W1;  // dpsi/ds2

  const float Gm1 = g1 * 0.5f / s1;                 // through sqrt
  const float Gm2 = g2 * 0.5f / s2;
  const float Ga1 = (a1 > kEps) ? Gm1 : 0.0f;       // through max(.,eps)
  const float Ga2 = (a2 > kEps) ? Gm2 : 0.0f;
  float Gtr   = 0.5f * (Ga1 + Ga2);
  const float Gdelta = 0.5f * (Ga1 - Ga2);
  const float Gdm = Gdelta * 0.5f / delta;          // through sqrt
  const float Gd2 = (d2 > kEps) ? Gdm : 0.0f;       // through max(.,eps)
  Gtr += 2.0f * tr * Gd2;
  const float Gdet = -4.0f * Gd2;

  // Gsym = Gc + Gc^T  (backward of C = F^T F gives dF = F * Gsym)
  const float Gs00 = 2.0f * (Gtr + Gdet * c11);
  const float Gs11 = 2.0f * (Gtr + Gdet * c00);
  const float Gs01 = -2.0f * Gdet * c01;

  const float P00 = Fn00 * Gs00 + Fn01 * Gs01;
  const float P01 = Fn00 * Gs01 + Fn01 * Gs11;
  const float P10 = Fn10 * Gs00 + Fn11 * Gs01;
  const float P11 = Fn10 * Gs01 + Fn11 * Gs11;

  // stress = kStressC * (P @ Fn^T);  affine = stress + P_MASS * C
  const float A00 = kStressC * (P00 * Fn00 + P01 * Fn01) + kPMass * C00;
  const float A01 = kStressC * (P00 * Fn10 + P01 * Fn11) + kPMass * C01;
  const float A10 = kStressC * (P10 * Fn00 + P11 * Fn01) + kPMass * C10;
  const float A11 = kStressC * (P10 * Fn10 + P11 * Fn11) + kPMass * C11;

  const float mv0 = kPMass * v0;
  const float mv1 = kPMass * v1;

  // Scatter: w * (m*v + A*(x_node - x_p)) and w*m  (== ref's mom/affine split + grid_a@grid_x)
  #pragma unroll
  for (int i = 0; i < 3; ++i) {
    const float dp0 = ((float)i - fx0) * kDX;
    const int gi = b0 + i;
    #pragma unroll
    for (int j = 0; j < 3; ++j) {
      const float dp1 = ((float)j - fx1) * kDX;
      const float w = wa[i] * wb[j];
      const int idx = gi * kGrid + (b1 + j);
      fatomic_add(&grid_v[2 * idx + 0], w * (mv0 + A00 * dp0 + A01 * dp1));
      fatomic_add(&grid_v[2 * idx + 1], w * (mv1 + A10 * dp0 + A11 * dp1));
      fatomic_add(&grid_m[idx], w * kPMass);
    }
  }

  // Streaming outputs: written once, never re-read this dispatch -> NT stores
  const v4f oF = {Fn00, Fn01, Fn10, Fn11};
  __builtin_nontemporal_store(oF, (v4f*)(out_F + 4 * (size_t)p));
  __builtin_nontemporal_store(matg[p], out_mat + p);
  __builtin_nontemporal_store(Jpg[p],  out_Jp + p);
}

__global__ __launch_bounds__(NPM_THREADS) void k_grid(float* __restrict__ gv,
                                                      const float* __restrict__ gm) {
  const int idx = blockIdx.x * NPM_THREADS + threadIdx.x;
  if (idx >= kCells) return;
  const int i = idx >> 8;
  const int j = idx & (kGrid - 1);
  const float m  = gm[idx];
  const float md = fmaxf(m, kEps);
  const bool ne  = m > kEps;
  float v0 = ne ? gv[2 * idx + 0] / md : 0.0f;
  float v1 = ne ? gv[2 * idx + 1] / md : 0.0f;
  v1 -= kDTG;                                  // gravity applied to all cells (matches ref)
  if (i < 3)         v0 = fmaxf(v0, 0.0f);
  if (i >= kGrid-3)  v0 = fminf(v0, 0.0f);
  if (j < 3)         v1 = fmaxf(v1, 0.0f);
  if (j >= kGrid-3)  v1 = fminf(v1, 0.0f);
  gv[2 * idx + 0] = v0;
  gv[2 * idx + 1] = v1;
}

__global__ __launch_bounds__(NPM_THREADS) void k_g2p(
    const float* __restrict__ xg, const float* __restrict__ vg,
    const float* __restrict__ gv,
    float* __restrict__ out_x, float* __restrict__ out_v, float* __restrict__ out_C,
    int n) {
  const int t = threadIdx.x;
  const int p = blockIdx.x * NPM_THREADS + t;

#if MPM_ASYNC_LDS
  __shared__ float sX[2 * NPM_THREADS];
  __shared__ float sV[2 * NPM_THREADS];
  if (p < n) {
    __builtin_amdgcn_global_load_async_to_lds_b64((GV2*)(xg + 2 * (size_t)p), (LV2*)(sX + 2 * t), 0, 0);
    __builtin_amdgcn_global_load_async_to_lds_b64((GV2*)(vg + 2 * (size_t)p), (LV2*)(sV + 2 * t), 0, 0);
  }
  wait_async0();
  if (p >= n) return;
  const float x0 = sX[2*t+0], x1 = sX[2*t+1];
  const float vv0 = sV[2*t+0], vv1 = sV[2*t+1];
#else
  if (p >= n) return;
  const float x0 = xg[2*p+0], x1 = xg[2*p+1];
  const float vv0 = vg[2*p+0], vv1 = vg[2*p+1];
#endif

  const int b0 = (int)(x0 * kInvDX - 0.5f);
  const int b1 = (int)(x1 * kInvDX - 0.5f);

  // gfx1250 global_prefetch_b8: pull the 3 grid rows of this particle's stencil
  {
    const char* pf = (const char*)(gv + 2 * (b0 * kGrid + b1));
    __builtin_prefetch(pf, 0, 1);
    __builtin_prefetch(pf + 2 * kGrid * sizeof(float), 0, 1);
    __builtin_prefetch(pf + 4 * kGrid * sizeof(float), 0, 1);
  }

  const float fx0 = x0 * kInvDX - (float)b0;
  const float fx1 = x1 * kInvDX - (float)b1;
  float wa[3], wb[3];
  wa[0] = 0.5f * (1.5f - fx0) * (1.5f - fx0);
  wa[1] = 0.75f - (fx0 - 1.0f) * (fx0 - 1.0f);
  wa[2] = 0.5f * (fx0 - 0.5f) * (fx0 - 0.5f);
  wb[0] = 0.5f * (1.5f - fx1) * (1.5f - fx1);
  wb[1] = 0.75f - (fx1 - 1.0f) * (fx1 - 1.0f);
  wb[2] = 0.5f * (fx1 - 0.5f) * (fx1 - 0.5f);

  float nv0 = 0.0f, nv1 = 0.0f;
  float c00 = 0.0f, c01 = 0.0f, c10 = 0.0f, c11 = 0.0f;
  #pragma unroll
  for (int i = 0; i < 3; ++i) {
    const float tp0 = (float)(b0 + i) * kDX;
    #pragma unroll
    for (int j = 0; j < 3; ++j) {
      const float w = wa[i] * wb[j];
      const int idx = (b0 + i) * kGrid + (b1 + j);
      const float g0 = gv[2 * idx + 0];
      const float g1 = gv[2 * idx + 1];
      const float tp1 = (float)(b1 + j) * kDX;
      nv0 += w * g0;
      nv1 += w * g1;
      c00 += w * g0 * tp0;  c01 += w * g0 * tp1;
      c10 += w * g1 * tp0;  c11 += w * g1 * tp1;
    }
  }
  c00 = (c00 - nv0 * x0) * kScaleC;
  c01 = (c01 - nv0 * x1) * kScaleC;
  c10 = (c10 - nv1 * x0) * kScaleC;
  c11 = (c11 - nv1 * x1) * kScaleC;

  const float nx0 = fminf(fmaxf(x0 + kDT * vv0, kXlo), kXhi);
  const float nx1 = fminf(fmaxf(x1 + kDT * vv1, kXlo), kXhi);

  const v2f ox = {nx0, nx1};
  const v2f ov = {nv0, nv1};
  const v4f oC = {c00, c01, c10, c11};
  __builtin_nontemporal_store(ox, (v2f*)(out_x + 2 * (size_t)p));
  __builtin_nontemporal_store(ov, (v2f*)(out_v + 2 * (size_t)p));
  __builtin_nontemporal_store(oC, (v4f*)(out_C + 4 * (size_t)p));
}

extern "C" void kernel_launch(void* const* d_in, const int* in_sizes, int n_in,
                              void* d_out, int out_size, void* d_ws, size_t ws_size,
                              hipStream_t stream) {
  (void)n_in; (void)out_size; (void)ws_size;
  const float* x   = (const float*)d_in[0];
  const float* v   = (const float*)d_in[1];
  const float* C   = (const float*)d_in[2];
  const float* F   = (const float*)d_in[3];
  const float* W   = (const float*)d_in[4];
  /* d_in[5] = b : additive constant in psi, zero gradient -> unused */
  const int*   mat = (const int*)d_in[6];
  const float* Jp  = (const float*)d_in[7];
  const int n = in_sizes[0] / 2;   // N_P

  // workspace: grid_v (2*kCells f32) + grid_m (kCells f32) = 768 KB
  float* gv = (float*)d_ws;
  float* gm = gv + 2 * (size_t)kCells;

  float* out_x = (float*)d_out;
  float* out_v = out_x + 2 * (size_t)n;
  float* out_C = out_v + 2 * (size_t)n;
  float* out_F = out_C + 4 * (size_t)n;
  int*   out_m = (int*)(out_F + 4 * (size_t)n);
  float* out_J = (float*)(out_m + n);

  const int c4 = (3 * kCells) / 4;                       // 49152 float4s
  const int zb = (c4 + NPM_THREADS - 1) / NPM_THREADS;
  const int pb = (n + NPM_THREADS - 1) / NPM_THREADS;

  k_zero4<<<zb, NPM_THREADS, 0, stream>>>((v4f*)gv, c4);
  k_p2g  <<<pb, NPM_THREADS, 0, stream>>>(x, v, C, F, W, mat, Jp,
                                          gv, gm, out_F, out_m, out_J, n);
  k_grid <<<kCells / NPM_THREADS, NPM_THREADS, 0, stream>>>(gv, gm);
  k_g2p  <<<pb, NPM_THREADS, 0, stream>>>(x, v, gv, out_x, out_v, out_C, n);
}